// AffineCoupling_21294447854290
// MI455X (gfx1250) — compile-verified
//
#include <hip/hip_runtime.h>
#include <math.h>

// ---------------------------------------------------------------------------
// AffineCoupling on MI455X (gfx1250), fp32 WMMA path.
//
// Per 16-row tile (one wave32):
//   A (16x16) = [logs-activation | b-activation]   (A operands: 16x4 f32, v2f)
//   B (16x16) = [Wl^T  0 ; 0  Wb^T] block diagonal (B operands: 4x16 f32, v2f)
//   D (16x16) = A*B + bias  via v_wmma_f32_16x16x4_f32 chains (f32 accum).
// Inter-layer D->A relayout via padded wave-private LDS tile (stride 20 f32).
// ---------------------------------------------------------------------------

typedef float v2f __attribute__((ext_vector_type(2)));
typedef float v4f __attribute__((ext_vector_type(4)));
typedef float v8f __attribute__((ext_vector_type(8)));

#define LDIM   16       // full row width (= 2*HALF)
#define TS     20       // padded LDS row stride in floats (gcd(20,64)=4 -> conflict-free, 16B aligned)
#define SLOPE  0.01f
#define WAVES_PER_BLOCK 4
#define THREADS (WAVES_PER_BLOCK * 32)

struct Params {
    const float* z;
    const float* wl[5];
    const float* bl[5];
    const float* wb[5];
    const float* bb[5];
    float* out;
    int ntiles;
};

__global__ __launch_bounds__(THREADS)
void affine_coupling_wmma(Params p) {
    const int lane = threadIdx.x & 31;
    const int wid  = threadIdx.x >> 5;
    const int n    = lane & 15;   // N column (B/C/D operand layout)
    const int h    = lane >> 4;   // lane half (K sub-block selector)

    // Two 16x20 f32 tiles per wave: Z = staged input rows, T = transpose scratch.
    __shared__ float smem[WAVES_PER_BLOCK * 2 * 16 * TS];
    float* Zt = smem + wid * (2 * 16 * TS);
    float* Tt = Zt + 16 * TS;

    // ---- build B operands (weights) in registers, once per wave ------------
    // B layout (4x16 slice): lane n = N column; VGPR j holds K = 4*s + 2*h + j.
    // Layer 1 (shared zl input, K=0..7): cols 0-7 <- Wl^T, cols 8-15 <- Wb^T.
    v2f B1[2];
    #pragma unroll
    for (int s = 0; s < 2; ++s)
        #pragma unroll
        for (int j = 0; j < 2; ++j) {
            const int k = 4 * s + 2 * h + j;
            B1[s][j] = (n < 8) ? p.wl[0][(n & 7) * 8 + k]
                               : p.wb[0][(n & 7) * 8 + k];
        }
    // Layers 2..5 (K=0..15): block-diagonal [Wl^T 0 ; 0 Wb^T].
    v2f BL[4][4];
    #pragma unroll
    for (int L = 0; L < 4; ++L)
        #pragma unroll
        for (int s = 0; s < 4; ++s)
            #pragma unroll
            for (int j = 0; j < 2; ++j) {
                const int k = 4 * s + 2 * h + j;
                float w;
                if (s < 2) w = (n < 8) ? p.wl[L + 1][(n & 7) * 8 + k] : 0.0f;
                else       w = (n < 8) ? 0.0f : p.wb[L + 1][(n & 7) * 8 + (k - 8)];
                BL[L][s][j] = w;
            }
    // Bias: in C/D layout every accumulator VGPR of a lane shares the same N,
    // so the per-layer bias is a single per-lane scalar.
    float bias[5];
    #pragma unroll
    for (int L = 0; L < 5; ++L)
        bias[L] = (n < 8) ? p.bl[L][n & 7] : p.bb[L][n & 7];

    const int wave_global = blockIdx.x * WAVES_PER_BLOCK + wid;
    const int wave_count  = gridDim.x * WAVES_PER_BLOCK;

    for (int tile = wave_global; tile < p.ntiles; tile += wave_count) {
        const int base = tile * 16;

        // ---- stage 16x16 z tile into LDS (coalesced b128) + zl passthrough -
        {
            const int r  = lane >> 1;
            const int cb = (lane & 1) * 8;
            const float* src = p.z + (size_t)(base + r) * LDIM + cb;
            v4f p0 = __builtin_nontemporal_load((const v4f*)src);
            v4f p1 = __builtin_nontemporal_load((const v4f*)(src + 4));
            *(v4f*)(Zt + r * TS + cb)     = p0;
            *(v4f*)(Zt + r * TS + cb + 4) = p1;
            if (cb == 0) {  // out[:, 0:8] = zl (identity half of the coupling)
                float* dst = p.out + (size_t)(base + r) * LDIM;
                __builtin_nontemporal_store(p0, (v4f*)dst);
                __builtin_nontemporal_store(p1, (v4f*)(dst + 4));
            }
        }
        // Same-wave DS ops are in order; compiler inserts s_wait_dscnt.

        // ---- layer 1: A from staged zl, K=8 --------------------------------
        v8f c;
        #pragma unroll
        for (int e = 0; e < 8; ++e) c[e] = bias[0];
        {
            // A layout: lane M = n, VGPR j holds K = 4*s + 2*h + j  -> b64 read
            v2f a0 = *(const v2f*)(Zt + n * TS + 2 * h);
            v2f a1 = *(const v2f*)(Zt + n * TS + 4 + 2 * h);
            c = __builtin_amdgcn_wmma_f32_16x16x4_f32(false, a0, false, B1[0],
                                                      (short)0, c, false, false);
            c = __builtin_amdgcn_wmma_f32_16x16x4_f32(false, a1, false, B1[1],
                                                      (short)0, c, false, false);
        }
        #pragma unroll
        for (int e = 0; e < 8; ++e) c[e] = fmaxf(c[e], SLOPE * c[e]);

        // ---- layers 2..5: transpose D->A via LDS, 4 WMMAs each -------------
        #pragma unroll
        for (int L = 0; L < 4; ++L) {
            #pragma unroll
            for (int v = 0; v < 8; ++v)            // D: reg v, lane -> (v+8h, n)
                Tt[(v + 8 * h) * TS + n] = c[v];
            v2f a[4];
            #pragma unroll
            for (int s = 0; s < 4; ++s)            // A: b64 -> K = 4s+2h+{0,1}
                a[s] = *(const v2f*)(Tt + n * TS + 4 * s + 2 * h);
            #pragma unroll
            for (int e = 0; e < 8; ++e) c[e] = bias[L + 1];
            #pragma unroll
            for (int s = 0; s < 4; ++s)
                c = __builtin_amdgcn_wmma_f32_16x16x4_f32(false, a[s], false,
                        BL[L][s], (short)0, c, false, false);
            if (L < 3) {
                #pragma unroll
                for (int e = 0; e < 8; ++e) c[e] = fmaxf(c[e], SLOPE * c[e]);
            }
        }

        // ---- epilogue: yr = exp(log_s) * zr + b ----------------------------
        #pragma unroll
        for (int v = 0; v < 8; ++v)
            Tt[(v + 8 * h) * TS + n] = c[v];       // cols 0-7 log_s, 8-15 b
        {
            const int m = n;          // tile row
            const int q = h;          // output column quarter: 4q .. 4q+3
            v4f ls = *(const v4f*)(Tt + m * TS + 4 * q);
            v4f bv = *(const v4f*)(Tt + m * TS + 8 + 4 * q);
            v4f zr = *(const v4f*)(Zt + m * TS + 8 + 4 * q);
            v4f y;
            #pragma unroll
            for (int e = 0; e < 4; ++e) y[e] = __expf(ls[e]) * zr[e] + bv[e];
            __builtin_nontemporal_store(
                y, (v4f*)(p.out + (size_t)(base + m) * LDIM + 8 + 4 * q));
        }
    }
}

extern "C" void kernel_launch(void* const* d_in, const int* in_sizes, int n_in,
                              void* d_out, int out_size, void* d_ws, size_t ws_size,
                              hipStream_t stream) {
    (void)n_in; (void)out_size; (void)d_ws; (void)ws_size;
    Params p;
    p.z = (const float*)d_in[0];
    for (int i = 0; i < 5; ++i) {
        p.wl[i] = (const float*)d_in[1 + i];    // ws_logs
        p.bl[i] = (const float*)d_in[6 + i];    // bs_logs
        p.wb[i] = (const float*)d_in[11 + i];   // ws_b
        p.bb[i] = (const float*)d_in[16 + i];   // bs_b
    }
    p.out = (float*)d_out;
    const int nrows = in_sizes[0] / LDIM;       // 4,000,000 (divisible by 16)
    p.ntiles = nrows / 16;

    const int blocks = 8192;                    // grid-stride persistent waves
    affine_coupling_wmma<<<blocks, THREADS, 0, stream>>>(p);
}